// SelfCorrelationComputation_1726576854246
// MI455X (gfx1250) — compile-verified
//
#include <hip/hip_runtime.h>
#include <math.h>

typedef float v2f __attribute__((ext_vector_type(2)));
typedef float v8f __attribute__((ext_vector_type(8)));

#define B_  16
#define C_  256
#define H_  32
#define W_  32
#define S_  (H_ * W_)          // 1024 spatial positions per (b)
#define NPIX (B_ * S_)         // 16384 pixels total
#define KW  5
#define PADR 2
#define TILE_ROWS (8 + 2 * PADR)   // 12 rows staged per block

// ---------------------------------------------------------------------------
// Kernel 1: inverse L2 norm over channels for each (b,h,w) pixel.
// Each wave32 handles 16 pixels. Per-lane f32 FMA partial sums of relu(x)^2
// over a disjoint channel subset (4-way split matching the A-matrix K slots),
// then ONE V_WMMA_F32_16X16X4_F32 with B = ones performs the cross-lane /
// cross-slot reduction: D[m,n] = sum_k A[m,k] = ||pixel m||^2.
// ---------------------------------------------------------------------------
__global__ __launch_bounds__(128) void norm_kernel(const float* __restrict__ x,
                                                   float* __restrict__ invn) {
  const int lane  = threadIdx.x & 31;
  const int wave  = blockIdx.x * 4 + (threadIdx.x >> 5);  // 4 waves / block
  const int pbase = wave * 16;                            // 16 pixels / wave
  const int b     = pbase >> 10;     // uniform across the wave (16 | 1024)
  const int s0    = pbase & 1023;
  const int p     = lane & 15;       // A-matrix row M = pixel within group
  const int kcls  = (lane >> 4) * 2; // K-slot pair: lanes<16 -> {0,1}, else {2,3}

  const float* xb = x + (size_t)b * C_ * S_ + (size_t)(s0 + p);

  v2f a; a.x = 0.0f; a.y = 0.0f;
  #pragma unroll 8
  for (int t = 0; t < 64; ++t) {
    const int c0 = 4 * t + kcls;
    float v0 = xb[(size_t)c0 * S_];
    float v1 = xb[(size_t)(c0 + 1) * S_];
    v0 = fmaxf(v0, 0.0f);
    v1 = fmaxf(v1, 0.0f);
    a.x = fmaf(v0, v0, a.x);
    a.y = fmaf(v1, v1, a.y);
  }

  v2f ones; ones.x = 1.0f; ones.y = 1.0f;
  v8f cacc = {};
  // D = A x ones + 0 : every column n of row m holds ||pixel m||^2
  v8f d = __builtin_amdgcn_wmma_f32_16x16x4_f32(
      /*neg_a=*/false, a, /*neg_b=*/false, ones,
      /*c_mod=*/(short)0, cacc, /*reuse_a=*/false, /*reuse_b=*/false);

  // D layout: VGPR r -> (lanes 0-15: M=r, N=lane) (lanes 16-31: M=r+8, N=lane-16)
  // lanes 0-7   hold pixel m = lane      at vgpr index lane
  // lanes 16-23 hold pixel m = lane - 8  at vgpr index lane - 16
  const float ss = ((const float*)&d)[lane & 7];
  const bool active = (lane < 8) || (lane >= 16 && lane < 24);
  if (active) {
    const int pix = (lane < 8) ? lane : (lane - 8);
    const float nrm = sqrtf(ss);
    invn[pbase + pix] = 1.0f / fmaxf(nrm, 1e-12f);
  }
}

// ---------------------------------------------------------------------------
// Kernel 2: streaming self-correlation, LDS-staged.
// One 256-thread block owns 8 rows (256 pixels) of one (b,c) plane.
//  a) stage zero-padded 12x32 tile of xn = relu(x)*invn in LDS
//  b) each thread computes its 25 products from LDS (bank-conflict-free,
//     stride-25 staging writes)
//  c) cooperative copy-out as 1600 aligned float4 -> contiguous
//     global_store_b128, store-bandwidth bound (~419 MB @ 23.3 TB/s ~ 18us)
// ---------------------------------------------------------------------------
__global__ __launch_bounds__(256) void corr_kernel(const float* __restrict__ x,
                                                   const float* __restrict__ invn,
                                                   float* __restrict__ out) {
  __shared__ alignas(16) float xn[TILE_ROWS * W_];      // 384 dwords
  __shared__ alignas(16) float stage[256 * KW * KW];    // 6400 dwords

  const int tid   = threadIdx.x;
  const int blk   = blockIdx.x;
  const int plane = blk >> 2;          // flat (b*256 + c), uniform per block
  const int b     = plane >> 8;
  const int s0    = (blk & 3) << 8;    // spatial base: 0/256/512/768
  const int h0    = s0 >> 5;           // row base (multiple of 8)
  const size_t xbase = (size_t)plane << 10;
  const int nbase = b << 10;

  // (a) stage normalized tile, rows h0-2 .. h0+9, zero rows outside the image
  for (int i = tid; i < TILE_ROWS * W_; i += 256) {
    const int r   = h0 - PADR + (i >> 5);
    const int col = i & 31;
    float v = 0.0f;
    if (r >= 0 && r < H_) {
      const int sp = (r << 5) | col;
      v = fmaxf(x[xbase + sp], 0.0f) * invn[nbase + sp];
    }
    xn[i] = v;
  }
  __syncthreads();

  // (b) 25 products per thread, entirely from LDS
  const int w    = tid & 31;
  const int lrow = (tid >> 5) + PADR;                // local row in tile
  const float center = xn[(lrow << 5) | w];

  float* myst = &stage[tid * (KW * KW)];
  #pragma unroll
  for (int u = 0; u < KW; ++u) {
    const int rr = (lrow + u - PADR) << 5;
    #pragma unroll
    for (int v = 0; v < KW; ++v) {
      const int ww = w + v - PADR;
      float rv = 0.0f;
      if (ww >= 0 && ww < W_) rv = center * xn[rr + ww];
      myst[u * KW + v] = rv;
    }
  }
  __syncthreads();

  // (c) contiguous vectorized copy-out: 6400 dwords = 1600 float4 per block
  const float4* st4 = (const float4*)stage;
  float4* o4 = (float4*)(out + (size_t)blk * (256 * KW * KW));
  for (int i = tid; i < 1600; i += 256) {
    o4[i] = st4[i];
  }
}

extern "C" void kernel_launch(void* const* d_in, const int* in_sizes, int n_in,
                              void* d_out, int out_size, void* d_ws, size_t ws_size,
                              hipStream_t stream) {
  (void)in_sizes; (void)n_in; (void)out_size; (void)ws_size;
  const float* x = (const float*)d_in[0];
  float* out = (float*)d_out;
  float* invn = (float*)d_ws;   // 16*1024 floats = 64 KB scratch

  // norm pass: 16384 pixels, 16 per wave, 4 waves per 128-thread block
  norm_kernel<<<NPIX / 64, 128, 0, stream>>>(x, invn);
  // main pass: 16384 blocks x 256 threads, one block per 8 rows of a plane
  corr_kernel<<<(B_ * C_ * S_) / 256, 256, 0, stream>>>(x, invn, out);
}